// RILayer_51513837748926
// MI455X (gfx1250) — compile-verified
//
#include <hip/hip_runtime.h>

#define D 128
#define CHUNK 256   // edges staged per block
#define EPW 32      // edges per wave (8 waves/block * 32 = 256)

typedef unsigned int u32x4 __attribute__((ext_vector_type(4)));
typedef int          i32x8 __attribute__((ext_vector_type(8)));
typedef int          i32x4 __attribute__((ext_vector_type(4)));

__device__ __forceinline__ void atomAddF(float* p, float v) {
    // non-returning f32 atomic add -> global_atomic_add_f32, STOREcnt path
    __hip_atomic_fetch_add(p, v, __ATOMIC_RELAXED, __HIP_MEMORY_SCOPE_AGENT);
}

__device__ __forceinline__ unsigned lds_byte_off(const void* p) {
    // flat pointer into LDS aperture: low 32 bits are the LDS byte offset
    return (unsigned)(unsigned long long)p;
}

// ---------------------------------------------------------------- r weights
__global__ void compute_r_kernel(const float* z, const float* f, const float* s,
                                 const float* t, float* r) {
    if (threadIdx.x == 0 && blockIdx.x == 0) {
        float a = fmaxf(z[0], 0.f), b = fmaxf(f[0], 0.f);
        float c = fmaxf(s[0], 0.f), d = fmaxf(t[0], 0.f);
        float inv = 1.0f / (a + b + c + d + 1e-6f);
        r[0] = a * inv; r[1] = b * inv; r[2] = c * inv; r[3] = d * inv;
    }
}

// ---------------------------------------------------------------- out = r0*x
__global__ __launch_bounds__(256) void init_out_kernel(const float* __restrict__ x,
                                                       const float* __restrict__ r,
                                                       float* __restrict__ out,
                                                       int nq) {
    int i = blockIdx.x * blockDim.x + threadIdx.x;
    if (i >= nq) return;
    float r0 = r[0];
    float4 v = ((const float4*)x)[i];
    v.x *= r0; v.y *= r0; v.z *= r0; v.w *= r0;
    ((float4*)out)[i] = v;
}

// ------------------------------------------- scatter: out[src,(d+roll)%128] += r*w*x[dst,d]
__global__ __launch_bounds__(256) void scatter_kernel(
    const long long* __restrict__ eidx,   // [2 * n_edges]: row0 = src, row1 = dst
    const float* __restrict__ w,
    const float* __restrict__ x,
    const float* __restrict__ rvec, int rk, int roll,
    float* __restrict__ out, int n_edges) {

    __shared__ long long sE[2 * CHUNK];   // [0..255]=src chunk, [256..511]=dst chunk
    __shared__ float     sW[CHUNK];

    const int chunk_base = blockIdx.x * CHUNK;

    if (threadIdx.x < 32) {
        // --- TDM: stage this block's edge slice into LDS asynchronously ---
        unsigned ldsE = lds_byte_off(sE);
        unsigned ldsW = lds_byte_off(sW);
        unsigned long long gaE =
            (unsigned long long)eidx + (unsigned long long)chunk_base * 8ull;
        unsigned long long gaW =
            (unsigned long long)w + (unsigned long long)chunk_base * 4ull;
        unsigned ne = (unsigned)n_edges;

        // D# group0: count=1 | lds_addr | global_addr[56:0] | type=2
        u32x4 g0e = { 1u, ldsE, (unsigned)gaE,
                      (unsigned)((gaE >> 32) & 0x01FFFFFFu) | (2u << 30) };
        // D# group1: 2D tile 256x2 of 8B elements, row stride = n_edges elems
        i32x8 g1e = { (int)(3u << 16),                                  // data_size=8B
                      (int)((ne & 0xFFFFu) << 16),                      // tensor_dim0 lo
                      (int)((ne >> 16) & 0xFFFFu) | (int)(2u << 16),    // dim0 hi | tensor_dim1=2
                      (int)(256u << 16),                                // tile_dim0=256
                      (int)2,                                           // tile_dim1=2
                      (int)ne, 0, 0 };                                  // dim0_stride=n_edges

        u32x4 g0w = { 1u, ldsW, (unsigned)gaW,
                      (unsigned)((gaW >> 32) & 0x01FFFFFFu) | (2u << 30) };
        // 1D tile of 256 f32
        i32x8 g1w = { (int)(2u << 16),                                  // data_size=4B
                      (int)((ne & 0xFFFFu) << 16),
                      (int)((ne >> 16) & 0xFFFFu) | (int)(1u << 16),
                      (int)(256u << 16),
                      (int)1,
                      (int)ne, 0, 0 };

        i32x4 z4 = { 0, 0, 0, 0 };
        i32x8 z8 = { 0, 0, 0, 0, 0, 0, 0, 0 };
        __builtin_amdgcn_tensor_load_to_lds(g0e, g1e, z4, z4, z8, 0);
        __builtin_amdgcn_tensor_load_to_lds(g0w, g1w, z4, z4, z8, 0);
        __builtin_amdgcn_s_wait_tensorcnt(0);
    }
    __syncthreads();

    const float rv  = rvec[rk];
    const int wave  = threadIdx.x >> 5;
    const int lane  = threadIdx.x & 31;

    for (int k = 0; k < EPW; ++k) {
        int el = wave * EPW + k;
        int eg = chunk_base + el;
        if (eg >= n_edges) break;

        // prefetch the gathered row 4 edges ahead; 32 lanes x 16B cover all 512B
        if (k + 4 < EPW) {
            long long dpf = sE[CHUNK + el + 4];
            __builtin_prefetch((const char*)(x + (size_t)dpf * D) + lane * 16, 0, 0);
        }

        long long src = sE[el];
        long long dst = sE[CHUNK + el];
        float cw = rv * sW[el];

        float4 v = ((const float4*)(x + (size_t)dst * D))[lane];   // global_load_b128
        float* orow = out + (size_t)src * D;
        int b = lane * 4 + roll;
        atomAddF(&orow[ b      & (D - 1)], cw * v.x);
        atomAddF(&orow[(b + 1) & (D - 1)], cw * v.y);
        atomAddF(&orow[(b + 2) & (D - 1)], cw * v.z);
        atomAddF(&orow[(b + 3) & (D - 1)], cw * v.w);
    }
}

// ---------------------------------------------------------------- launcher
extern "C" void kernel_launch(void* const* d_in, const int* in_sizes, int n_in,
                              void* d_out, int out_size, void* d_ws, size_t ws_size,
                              hipStream_t stream) {
    const float*     x  = (const float*)d_in[0];
    const long long* e1 = (const long long*)d_in[1];
    const float*     w1 = (const float*)d_in[2];
    const long long* e2 = (const long long*)d_in[3];
    const float*     w2 = (const float*)d_in[4];
    const long long* e3 = (const long long*)d_in[5];
    const float*     w3 = (const float*)d_in[6];
    const float*     z  = (const float*)d_in[7];
    const float*     f  = (const float*)d_in[8];
    const float*     s  = (const float*)d_in[9];
    const float*     t  = (const float*)d_in[10];

    float* out = (float*)d_out;
    float* r   = (float*)d_ws;   // 4 floats

    int nE1 = in_sizes[2], nE2 = in_sizes[4], nE3 = in_sizes[6];
    int nq  = in_sizes[0] / 4;   // float4 count of x / out

    compute_r_kernel<<<1, 32, 0, stream>>>(z, f, s, t, r);
    init_out_kernel<<<(nq + 255) / 256, 256, 0, stream>>>(x, r, out, nq);

    scatter_kernel<<<(nE1 + CHUNK - 1) / CHUNK, 256, 0, stream>>>(e1, w1, x, r, 1, 0, out, nE1);
    scatter_kernel<<<(nE2 + CHUNK - 1) / CHUNK, 256, 0, stream>>>(e2, w2, x, r, 2, 1, out, nE2);
    scatter_kernel<<<(nE3 + CHUNK - 1) / CHUNK, 256, 0, stream>>>(e3, w3, x, r, 3, 2, out, nE3);
}